// BasicWHVILinear_4363686772935
// MI455X (gfx1250) — compile-verified
//
#include <hip/hip_runtime.h>
#include <hip/hip_bf16.h>

#define DD   2048
#define BATCH 16384
#define NB   128            // Cholesky block size
#define INV_SQRT_D 0.02209708691207961f   // 1/sqrt(2048)

typedef __attribute__((ext_vector_type(2))) float v2f;
typedef __attribute__((ext_vector_type(8))) float v8f;

// ---------------------------------------------------------------------------
// qf = lower + lower^T with diagonal = diag(lower)
// ---------------------------------------------------------------------------
__global__ void build_qf_kernel(const float* __restrict__ low, float* __restrict__ qf) {
    int idx = blockIdx.x * 256 + threadIdx.x;          // idx < D*D
    int i = idx >> 11, j = idx & (DD - 1);
    float v = (i == j) ? low[idx] : (low[(size_t)i * DD + j] + low[(size_t)j * DD + i]);
    qf[idx] = v;
}

// ---------------------------------------------------------------------------
// WMMA SYRK:  C[r][c] (r,c in [0,n)) {=, -=} sum_{t<K} A[r][t]*A[c][t]
// One 16x16 C tile per wave; V_WMMA_F32_16X16X4_F32, K-step = 4 (unrolls to 32).
// ---------------------------------------------------------------------------
__global__ void wmma_syrk_kernel(float* __restrict__ C, int ldc,
                                 const float* __restrict__ A, int lda,
                                 int n, int K, int subtract) {
    int wave = (blockIdx.x * 256 + threadIdx.x) >> 5;  // 8 waves / block
    int lane = threadIdx.x & 31;
    int nt = n >> 4;
    if (wave >= nt * nt) return;                        // wave-uniform exit (EXEC all-1 for WMMA)
    int ti = wave / nt, tj = wave % nt;
    int ml = lane & 15, lh = lane >> 4;

    const float* arow = A + (size_t)(ti * 16 + ml) * lda;
    const float* brow = A + (size_t)(tj * 16 + ml) * lda;

    v8f acc = {};
    for (int k = 0; k < K; k += 4) {
        v2f a = *(const v2f*)(arow + k + 2 * lh);
        v2f b = *(const v2f*)(brow + k + 2 * lh);
        acc = __builtin_amdgcn_wmma_f32_16x16x4_f32(
            /*neg_a=*/false, a, /*neg_b=*/false, b,
            /*c_mod=*/(short)0, acc, /*reuse_a=*/false, /*reuse_b=*/false);
    }
    // C/D layout: VGPR r -> row (8*lh + r), column ml
    int rbase = ti * 16 + lh * 8;
    int c0 = tj * 16 + ml;
    #pragma unroll
    for (int r = 0; r < 8; ++r) {
        float* cp = C + (size_t)(rbase + r) * ldc + c0;
        if (subtract) *cp -= acc[r];
        else          *cp  = acc[r];
    }
}

// ---------------------------------------------------------------------------
// Cholesky diagonal block factor: Sig[s:s+NB, s:s+NB] lower <- chol, in LDS.
// ---------------------------------------------------------------------------
__global__ void chol_diag_kernel(float* __restrict__ Sig, int ld, int s) {
    __shared__ float T[NB][NB];                         // 64 KB
    float* base = Sig + (size_t)s * ld + s;
    int t = threadIdx.x;                                // 256 threads
    for (int idx = t; idx < NB * NB; idx += 256) {
        int i = idx >> 7, j = idx & (NB - 1);
        T[i][j] = base[(size_t)i * ld + j];
    }
    __syncthreads();
    for (int k = 0; k < NB; ++k) {
        if (t == 0) T[k][k] = sqrtf(T[k][k]);
        __syncthreads();
        float d = T[k][k];
        for (int i = k + 1 + t; i < NB; i += 256) T[i][k] /= d;
        __syncthreads();
        int rem = NB - k - 1;
        for (int idx = t; idx < rem * rem; idx += 256) {
            int i = k + 1 + idx / rem;
            int j = k + 1 + idx % rem;
            T[i][j] -= T[i][k] * T[j][k];               // full trailing square; upper junk harmless
        }
        __syncthreads();
    }
    for (int idx = t; idx < NB * NB; idx += 256) {
        int i = idx >> 7, j = idx & (NB - 1);
        if (j <= i) base[(size_t)i * ld + j] = T[i][j];
    }
}

// ---------------------------------------------------------------------------
// TRSM: L21 = A21 * L11^{-T}. One row per thread (64 threads/block).
// Per-row solution vector kept in LDS (xs[j][t]) -> conflict-free, no spills.
// ---------------------------------------------------------------------------
__global__ void chol_trsm_kernel(float* __restrict__ Sig, int ld, int s, int nrows) {
    __shared__ float Lb[NB][NB];                        // 64 KB
    __shared__ float xs[NB][64];                        // 32 KB
    int t = threadIdx.x;                                // 64 threads
    const float* diag = Sig + (size_t)s * ld + s;
    for (int idx = t; idx < NB * NB; idx += 64) {
        int i = idx >> 7, j = idx & (NB - 1);
        Lb[i][j] = diag[(size_t)i * ld + j];
    }
    __syncthreads();
    int r = blockIdx.x * 64 + t;
    if (r >= nrows) return;
    float* row = Sig + (size_t)(s + NB + r) * ld + s;
    for (int j = 0; j < NB; ++j) xs[j][t] = row[j];
    for (int j = 0; j < NB; ++j) {
        float v = xs[j][t];
        for (int p = 0; p < j; ++p) v -= xs[p][t] * Lb[j][p];
        xs[j][t] = v / Lb[j][j];
    }
    for (int j = 0; j < NB; ++j) row[j] = xs[j][t];
}

// ---------------------------------------------------------------------------
// g = q_mu + L * eps   (lower-triangular matvec; L in lower triangle of Sig)
// ---------------------------------------------------------------------------
__global__ void matvec_g_kernel(const float* __restrict__ L, int ld,
                                const float* __restrict__ qmu,
                                const float* __restrict__ eps,
                                float* __restrict__ g) {
    int i = blockIdx.x * 256 + threadIdx.x;
    if (i >= DD) return;
    const float* row = L + (size_t)i * ld;
    float acc = 0.f;
    for (int j = 0; j <= i; ++j) acc += row[j] * eps[j];
    g[i] = qmu[i] + acc;
}

// ---------------------------------------------------------------------------
// u = (1/sqrt(D)) * FWHT(s1 * g)   — Sylvester-order butterfly == reference H
// ---------------------------------------------------------------------------
__global__ void fwht_u_kernel(const float* __restrict__ g, const float* __restrict__ s1,
                              float* __restrict__ u) {
    __shared__ float v[DD];
    int t = threadIdx.x;                                // 1024 threads
    v[t]        = s1[t] * g[t];
    v[t + 1024] = s1[t + 1024] * g[t + 1024];
    __syncthreads();
    for (int h = 1; h < DD; h <<= 1) {
        int i = ((t & ~(h - 1)) << 1) | (t & (h - 1));
        float a = v[i], b = v[i + h];
        v[i] = a + b; v[i + h] = a - b;
        __syncthreads();
    }
    u[t]        = v[t] * INV_SQRT_D;
    u[t + 1024] = v[t + 1024] * INV_SQRT_D;
}

// ---------------------------------------------------------------------------
// Heavy kernel: out[b,:] = relu(s2 * (1/sqrt(D)) * FWHT(x[b,:] * u))
// One row per 256-thread block, 8 KB LDS.
// Row load:  GLOBAL_LOAD_ASYNC_TO_LDS_B128   (ASYNCcnt, no VGPR staging)
// Row store: GLOBAL_STORE_ASYNC_FROM_LDS_B128 (drained by S_ENDPGM wait-idle)
// ---------------------------------------------------------------------------
__global__ void fwht_rows_kernel(const float* __restrict__ x, const float* __restrict__ u,
                                 const float* __restrict__ s2, float* __restrict__ out) {
    __shared__ float v[DD];
    const int t = threadIdx.x;
    const size_t row = blockIdx.x;

    // --- async copy: 8 KB row -> LDS, 2 x b128 per thread (offset hits both sides)
    {
        const float* g0 = x + row * DD + 4 * t;
        unsigned l0 = (unsigned)(uintptr_t)(&v[4 * t]);     // LDS byte address = low 32 bits
        asm volatile("global_load_async_to_lds_b128 %0, %1, off"
                     :: "v"(l0), "v"(g0) : "memory");
        asm volatile("global_load_async_to_lds_b128 %0, %1, off offset:4096"
                     :: "v"(l0), "v"(g0) : "memory");
        asm volatile("s_wait_asynccnt 0x0" ::: "memory");
    }
    __syncthreads();

    // --- elementwise multiply by u (in place)
    for (int i = t; i < DD; i += 256) v[i] *= u[i];
    __syncthreads();

    // --- 11 butterfly stages (Sylvester order == reference Hadamard)
    for (int h = 1; h < DD; h <<= 1) {
        #pragma unroll
        for (int q = 0; q < 4; ++q) {
            int p = t + q * 256;                        // 1024 disjoint pairs / stage
            int i = ((p & ~(h - 1)) << 1) | (p & (h - 1));
            float a = v[i], b = v[i + h];
            v[i] = a + b; v[i + h] = a - b;
        }
        __syncthreads();
    }

    // --- scale * s2, relu, write back into LDS
    for (int i = t; i < DD; i += 256) v[i] = fmaxf(v[i] * s2[i] * INV_SQRT_D, 0.f);
    __syncthreads();

    // --- async store: LDS -> 8 KB output row
    {
        float* o0 = out + row * DD + 4 * t;
        unsigned l0 = (unsigned)(uintptr_t)(&v[4 * t]);
        asm volatile("global_store_async_from_lds_b128 %0, %1, off"
                     :: "v"(o0), "v"(l0) : "memory");
        asm volatile("global_store_async_from_lds_b128 %0, %1, off offset:4096"
                     :: "v"(o0), "v"(l0) : "memory");
        // S_ENDPGM performs an implicit wait-idle; no explicit s_wait needed.
    }
}

// ---------------------------------------------------------------------------
extern "C" void kernel_launch(void* const* d_in, const int* in_sizes, int n_in,
                              void* d_out, int out_size, void* d_ws, size_t ws_size,
                              hipStream_t stream) {
    const float* x   = (const float*)d_in[0];
    const float* s1  = (const float*)d_in[1];
    const float* s2  = (const float*)d_in[2];
    const float* qmu = (const float*)d_in[3];
    const float* qfl = (const float*)d_in[4];
    const float* eps = (const float*)d_in[5];
    float* out = (float*)d_out;

    // workspace layout: qf (D*D) | Sig/L (D*D) | g (D) | u (D)  = ~33.6 MB
    float* qf  = (float*)d_ws;
    float* Sig = qf + (size_t)DD * DD;
    float* g   = Sig + (size_t)DD * DD;
    float* u   = g + DD;

    // 1) qf
    build_qf_kernel<<<(DD * DD) / 256, 256, 0, stream>>>(qfl, qf);

    // 2) Sigma = qf * qf^T  (WMMA SYRK, full 2048^2, K=2048)
    {
        int nt = DD / 16, waves = nt * nt;
        wmma_syrk_kernel<<<(waves + 7) / 8, 256, 0, stream>>>(Sig, DD, qf, DD, DD, DD, 0);
    }

    // 3) blocked Cholesky of Sigma (in place, lower triangle)
    for (int kb = 0; kb < DD / NB; ++kb) {
        int s = kb * NB;
        chol_diag_kernel<<<1, 256, 0, stream>>>(Sig, DD, s);
        int rem = DD - s - NB;
        if (rem > 0) {
            chol_trsm_kernel<<<(rem + 63) / 64, 64, 0, stream>>>(Sig, DD, s, rem);
            int nt = rem / 16, waves = nt * nt;
            wmma_syrk_kernel<<<(waves + 7) / 8, 256, 0, stream>>>(
                Sig + (size_t)(s + NB) * DD + (s + NB), DD,       // C = trailing block
                Sig + (size_t)(s + NB) * DD + s, DD,              // A = L21 panel
                rem, NB, 1);
        }
    }

    // 4) g = q_mu + L*eps
    matvec_g_kernel<<<DD / 256, 256, 0, stream>>>(Sig, DD, qmu, eps, g);

    // 5) u = H*(s1*g)  via FWHT
    fwht_u_kernel<<<1, 1024, 0, stream>>>(g, s1, u);

    // 6) out = relu(s2 * H * (x ⊙ u))  — bandwidth-bound FWHT per row, async LDS path
    fwht_rows_kernel<<<BATCH, 256, 0, stream>>>(x, u, s2, out);
}